// Cropper_66451734003980
// MI455X (gfx1250) — compile-verified
//
#include <hip/hip_runtime.h>
#include <hip/hip_bf16.h>

// Problem constants (match the reference)
#define BB    8
#define NBOXK 100
#define CC    3
#define HH    1024
#define WW    1024
#define SS    100   // CROP

// Grid: (NBOX, S, B). One block per (box, output row); 128 threads = 4 wave32s,
// lane j = output column. All 12 bilinear gathers per thread issue before one wait.
__global__ __launch_bounds__(128) void Cropper_66451734003980_kernel(
    const float* __restrict__ images,   // (B, C, H, W) fp32
    const float* __restrict__ boxes,    // (B*NBOX, 4)  fp32 [bx, by, bw, bh]
    float* __restrict__ out)            // (B*NBOX, C, S, S) fp32
{
    const int b  = blockIdx.z;                  // image index 0..B-1
    const int bn = b * NBOXK + blockIdx.x;      // flat box index 0..B*NBOX-1
    const int i  = blockIdx.y;                  // output row 0..S-1
    const int j  = threadIdx.x;                 // output col
    if (j >= SS) return;

    // Per-box parameters: uniform 16B load (compiler lowers to s_load_b128).
    const float4 bx4 = ((const float4*)boxes)[bn];
    const float x0 = floorf(bx4.x * (float)WW);
    const float y0 = floorf(bx4.y * (float)HH);
    const float w0 = fmaxf(floorf(bx4.z * (float)WW), 1.0f);
    const float h0 = fmaxf(floorf(bx4.w * (float)HH), 1.0f);

    // ---- y tap for this output row (IEEE-exact division, matches reference) ----
    const float gy = ((float)i + 0.5f) / (float)SS;
    const float sy = fminf(fmaxf(gy * h0 - 0.5f, 0.0f), h0 - 1.0f) + y0;
    const float yf = floorf(sy);
    const float wy = sy - yf;
    const int y0i = (int)fminf(fmaxf(yf,        0.0f), (float)(HH - 1));
    const int y1i = (int)fminf(fmaxf(yf + 1.0f, 0.0f), (float)(HH - 1));

    // ---- x tap for this output column ----
    const float gx = ((float)j + 0.5f) / (float)SS;
    const float sx = fminf(fmaxf(gx * w0 - 0.5f, 0.0f), w0 - 1.0f) + x0;
    const float xf = floorf(sx);
    const float wx = sx - xf;
    const int x0i = (int)fminf(fmaxf(xf,        0.0f), (float)(WW - 1));
    const int x1i = (int)fminf(fmaxf(xf + 1.0f, 0.0f), (float)(WW - 1));

    const float omwx = 1.0f - wx;
    const float omwy = 1.0f - wy;

    const size_t plane   = (size_t)HH * (size_t)WW;          // 1 Mi elements
    const size_t imgBase = (size_t)b * (size_t)CC * plane;
    const float* __restrict__ row0 = images + imgBase + (size_t)y0i * (size_t)WW;
    const float* __restrict__ row1 = images + imgBase + (size_t)y1i * (size_t)WW;

    // Warm lines for channel-1/2 gathers while channel-0 loads are in flight.
    // (gfx1250: lowers to global_prefetch_b8)
    __builtin_prefetch(row0 + plane     + x0i, 0, 2);
    __builtin_prefetch(row1 + plane     + x0i, 0, 2);
    __builtin_prefetch(row0 + 2 * plane + x0i, 0, 2);
    __builtin_prefetch(row1 + 2 * plane + x0i, 0, 2);

    const size_t obase = (size_t)bn * (size_t)(CC * SS * SS)
                       + (size_t)i * (size_t)SS + (size_t)j;

    #pragma unroll
    for (int c = 0; c < CC; ++c) {
        const float* __restrict__ r0 = row0 + (size_t)c * plane;
        const float* __restrict__ r1 = row1 + (size_t)c * plane;
        const float v00 = r0[x0i];
        const float v01 = r0[x1i];
        const float v10 = r1[x0i];
        const float v11 = r1[x1i];
        // Same 4-term expression as the reference for numerical fidelity.
        const float o = v00 * omwy * omwx
                      + v01 * omwy * wx
                      + v10 * wy   * omwx
                      + v11 * wy   * wx;
        out[obase + (size_t)c * (size_t)(SS * SS)] = o;
    }
}

extern "C" void kernel_launch(void* const* d_in, const int* in_sizes, int n_in,
                              void* d_out, int out_size, void* d_ws, size_t ws_size,
                              hipStream_t stream) {
    (void)in_sizes; (void)n_in; (void)d_ws; (void)ws_size; (void)out_size;
    const float* images = (const float*)d_in[0];  // 8*3*1024*1024 fp32
    const float* boxes  = (const float*)d_in[1];  // 8*100*4 fp32
    float* out          = (float*)d_out;          // 800*3*100*100 fp32

    dim3 grid(NBOXK, SS, BB);    // (100, 100, 8) blocks = 80,000
    dim3 block(128);             // 4 wave32s; lanes 0..99 active
    Cropper_66451734003980_kernel<<<grid, block, 0, stream>>>(images, boxes, out);
}